// RoiPoolingConv_1683627180057
// MI455X (gfx1250) — compile-verified
//
#include <hip/hip_runtime.h>

// ROI bilinear pool: img [1,128,128,1024] f32 (NHWC), rois [1,512,4] i32 (x,y,w,h)
// out [1,512,14,14,1024] f32.
// Memory-bound: ~411MB NT output stream + 64MB L2-resident image gathers.

#define POOL   14
#define IMG_W  128
#define IMG_C  1024

typedef float v4f __attribute__((ext_vector_type(4)));
typedef int   v4i __attribute__((ext_vector_type(4)));

#if __has_builtin(__builtin_amdgcn_global_load_async_to_lds_b128) && \
    __has_builtin(__builtin_amdgcn_s_wait_asynccnt)
#define USE_ASYNC_LDS 1
#define AS1 __attribute__((address_space(1)))
#define AS3 __attribute__((address_space(3)))
#else
#define USE_ASYNC_LDS 0
#endif

__device__ __forceinline__ float lerpf(float a, float b, float t) {
    // a*(1-t) + b*t
    return __builtin_fmaf(t, b, __builtin_fmaf(-t, a, a));
}

__global__ __launch_bounds__(256)
void roi_bilinear_kernel(const float* __restrict__ img,
                         const int*   __restrict__ rois,
                         float*       __restrict__ out)
{
#if USE_ASYNC_LDS
    __shared__ float lds[2][4][IMG_C];   // 2 pipeline stages x 4 src pixels x 4KB = 32KB
#endif
    const int py  = blockIdx.x;          // output row 0..13
    const int roi = blockIdx.y;          // roi index
    const int tid = threadIdx.x;         // 256 threads
    const int c   = tid * 4;             // float4 channel slice: 256*4 = 1024 = IMG_C

    // Block-uniform ROI math -> SALU / s_load.
    const int rx = rois[roi * 4 + 0];
    const int ry = rois[roi * 4 + 1];
    const int rw = rois[roi * 4 + 2];
    const int rh = rois[roi * 4 + 3];

    const float fx = (float)rw / (float)POOL;
    const float fy = (float)rh / (float)POOL;

    // Vertical interpolation setup for this output row.
    const float gy  = (float)ry + (float)py * fy;
    const int   y0  = (int)floorf(gy);
    const float wy  = gy - (float)y0;
    const int   yhi = ry + rh - 1;
    const int   y0c = min(max(y0,     ry), yhi);
    const int   y1c = min(max(y0 + 1, ry), yhi);

    float* const orow = out + ((long long)(roi * (POOL * POOL) + py * POOL)) * IMG_C;

    // Per-output-pixel source offsets (element offsets incl. channel slice).
    auto calc = [&](int px, int o[4]) -> float {
        const float gx  = (float)rx + (float)px * fx;
        const int   x0  = (int)floorf(gx);
        const float wx  = gx - (float)x0;
        const int   xhi = rx + rw - 1;
        const int   x0c = min(max(x0,     rx), xhi);
        const int   x1c = min(max(x0 + 1, rx), xhi);
        o[0] = (y0c * IMG_W + x0c) * IMG_C + c;
        o[1] = (y0c * IMG_W + x1c) * IMG_C + c;
        o[2] = (y1c * IMG_W + x0c) * IMG_C + c;
        o[3] = (y1c * IMG_W + x1c) * IMG_C + c;
        return wx;
    };

    auto blend_store = [&](int px, float wx, v4f v00, v4f v01, v4f v10, v4f v11) {
        v4f res;
        #pragma unroll
        for (int k = 0; k < 4; ++k) {
            const float top = lerpf(v00[k], v01[k], wx);
            const float bot = lerpf(v10[k], v11[k], wx);
            res[k] = lerpf(top, bot, wy);
        }
        // Write-once 411MB stream: non-temporal so the 64MB image stays L2-resident.
        __builtin_nontemporal_store(res, (v4f*)(orow + px * IMG_C + c));
    };

#if USE_ASYNC_LDS
    // CDNA5 async global->LDS DMA, double-buffered across the 14 output pixels
    // of this row. Each stage: 4 async b128 gathers per lane (16KB / stage).
    // Builtin signature (from probe): (v4i addrspace(1)* src, v4i addrspace(3)* dst,
    //                                  imm offset, imm cpol)
    auto issue = [&](int buf, const int o[4]) {
        #pragma unroll
        for (int s = 0; s < 4; ++s) {
            __builtin_amdgcn_global_load_async_to_lds_b128(
                (AS1 v4i*)(img + o[s]),
                (AS3 v4i*)&lds[buf][s][c],
                0, 0);
        }
    };

    int   offs[2][4];
    float wxv[2];

    wxv[0] = calc(0, offs[0]);
    issue(0, offs[0]);

    int buf = 0;
    for (int px = 0; px < POOL - 1; ++px) {
        // Prefetch next pixel's 4 source vectors into the other buffer.
        wxv[buf ^ 1] = calc(px + 1, offs[buf ^ 1]);
        issue(buf ^ 1, offs[buf ^ 1]);
        // 8 async ops outstanding; <=4 left means current stage's 4 are done
        // (async loads complete in order).
        __builtin_amdgcn_s_wait_asynccnt(4);
        const v4f v00 = *(const v4f*)&lds[buf][0][c];
        const v4f v01 = *(const v4f*)&lds[buf][1][c];
        const v4f v10 = *(const v4f*)&lds[buf][2][c];
        const v4f v11 = *(const v4f*)&lds[buf][3][c];
        blend_store(px, wxv[buf], v00, v01, v10, v11);
        buf ^= 1;
    }
    __builtin_amdgcn_s_wait_asynccnt(0);
    {
        const v4f v00 = *(const v4f*)&lds[buf][0][c];
        const v4f v01 = *(const v4f*)&lds[buf][1][c];
        const v4f v10 = *(const v4f*)&lds[buf][2][c];
        const v4f v11 = *(const v4f*)&lds[buf][3][c];
        blend_store(POOL - 1, wxv[buf], v00, v01, v10, v11);
    }
#else
    // Fallback: direct b128 gathers (L2-resident image), latency hidden by occupancy.
    for (int px = 0; px < POOL; ++px) {
        int o[4];
        const float wx  = calc(px, o);
        const v4f   v00 = *(const v4f*)(img + o[0]);
        const v4f   v01 = *(const v4f*)(img + o[1]);
        const v4f   v10 = *(const v4f*)(img + o[2]);
        const v4f   v11 = *(const v4f*)(img + o[3]);
        blend_store(px, wx, v00, v01, v10, v11);
    }
#endif
}

extern "C" void kernel_launch(void* const* d_in, const int* in_sizes, int n_in,
                              void* d_out, int out_size, void* d_ws, size_t ws_size,
                              hipStream_t stream) {
    const float* img  = (const float*)d_in[0];
    const int*   rois = (const int*)d_in[1];
    float*       out  = (float*)d_out;

    const int nroi = in_sizes[1] / 4;   // 512
    dim3 grid(POOL, nroi);              // (output row, roi)
    roi_bilinear_kernel<<<grid, 256, 0, stream>>>(img, rois, out);

    (void)n_in; (void)out_size; (void)d_ws; (void)ws_size;
}